// CoxSAGE_64355789963692
// MI455X (gfx1250) — compile-verified
//
#include <hip/hip_runtime.h>
#include <hip/hip_bf16.h>
#include <math.h>

typedef float v2f __attribute__((ext_vector_type(2)));
typedef float v8f __attribute__((ext_vector_type(8)));

#define H 64

__device__ __forceinline__ float elu1(float x) {
    return x > 0.0f ? x : (__expf(x) - 1.0f);
}

// ---------------- encoders: h = elu(x @ W + b), small K ----------------
__global__ void enc_kernel(const float* __restrict__ x, const float* __restrict__ W,
                           const float* __restrict__ b, float* __restrict__ h,
                           int N, int F) {
    int t = blockIdx.x * blockDim.x + threadIdx.x;
    int n = t >> 6, j = t & 63;
    if (n >= N) return;
    const float* xr = x + (long)n * F;
    float s = b[j];
    for (int k = 0; k < F; ++k) s = fmaf(xr[k], W[k * H + j], s);
    h[(long)n * H + j] = elu1(s);
}

// ---------------- scatter-sum of messages + degree count ----------------
__global__ void scatter_kernel(const float* __restrict__ hsrc, const int* __restrict__ src,
                               const int* __restrict__ dst, float* __restrict__ agg,
                               float* __restrict__ deg, int E) {
    long t = (long)blockIdx.x * blockDim.x + threadIdx.x;
    int e = (int)(t >> 6), j = (int)(t & 63);
    if (e >= E) return;
    int s = src[e], d = dst[e];
    atomicAdd(&agg[(long)d * H + j], hsrc[(long)s * H + j]);
    if (j == 0) atomicAdd(&deg[d], 1.0f);
}

// ---------------- SAGEConv GEMM via V_WMMA_F32_16X16X4_F32 ----------------
// hnew[n][:] += (agg[n]/max(deg,1)) @ Wl + bl + hdst[n] @ Wr
// one wave -> 16x64 output tile; 4 waves per block share weights in LDS.
__global__ void __launch_bounds__(128)
sage_gemm_kernel(const float* __restrict__ agg, const float* __restrict__ deg,
                 const float* __restrict__ hdst,
                 const float* __restrict__ Wl, const float* __restrict__ bl,
                 const float* __restrict__ Wr,
                 float* __restrict__ hnew, int N) {
    __shared__ float sWl[H * H];
    __shared__ float sWr[H * H];
    int tid = threadIdx.x;
    {
        const float4* gl = (const float4*)Wl;
        const float4* gr = (const float4*)Wr;
        float4* ll = (float4*)sWl;
        float4* lr = (float4*)sWr;
        for (int i = tid; i < (H * H / 4); i += 128) {
            ll[i] = gl[i];
            lr[i] = gr[i];
        }
    }
    __syncthreads();

    int wave = tid >> 5;
    int lane = tid & 31;
    int hi   = lane >> 4;          // 0 or 1 (half-wave)
    int col  = lane & 15;
    int kk   = hi * 2;             // f32 fragment K sub-offset
    int m0   = blockIdx.x * 64 + wave * 16;

    int r  = m0 + col;             // A-fragment row for this lane
    int rc = r < N ? r : (N - 1);
    float invd = 1.0f / fmaxf(deg[rc], 1.0f);

    const float* aggr = agg  + (long)rc * H;
    const float* hdr  = hdst + (long)rc * H;

    v8f acc[4] = {};

    // pass 1: (agg * 1/deg) @ Wl
#pragma unroll
    for (int k0 = 0; k0 < H; k0 += 4) {
        v2f a;
        a.x = aggr[k0 + kk] * invd;
        a.y = aggr[k0 + kk + 1] * invd;
#pragma unroll
        for (int t4 = 0; t4 < 4; ++t4) {
            v2f b;
            b.x = sWl[(k0 + kk) * H + t4 * 16 + col];
            b.y = sWl[(k0 + kk + 1) * H + t4 * 16 + col];
            acc[t4] = __builtin_amdgcn_wmma_f32_16x16x4_f32(
                false, a, false, b, (short)0, acc[t4], false, false);
        }
    }
    // pass 2: hdst @ Wr
#pragma unroll
    for (int k0 = 0; k0 < H; k0 += 4) {
        v2f a;
        a.x = hdr[k0 + kk];
        a.y = hdr[k0 + kk + 1];
#pragma unroll
        for (int t4 = 0; t4 < 4; ++t4) {
            v2f b;
            b.x = sWr[(k0 + kk) * H + t4 * 16 + col];
            b.y = sWr[(k0 + kk + 1) * H + t4 * 16 + col];
            acc[t4] = __builtin_amdgcn_wmma_f32_16x16x4_f32(
                false, a, false, b, (short)0, acc[t4], false, false);
        }
    }

    // epilogue: D layout -> VGPR rr: lanes0-15 M=rr, lanes16-31 M=rr+8, N=col
#pragma unroll
    for (int t4 = 0; t4 < 4; ++t4) {
        int cg = t4 * 16 + col;
        float bias = bl[cg];
#pragma unroll
        for (int rr = 0; rr < 8; ++rr) {
            int row = m0 + rr + hi * 8;
            if (row < N) hnew[(long)row * H + cg] += acc[t4][rr] + bias;
        }
    }
}

// ---------------- residual + LayerNorm + ELU (one wave32 per node) ----------------
__global__ void ln_kernel(float* __restrict__ h, const float* __restrict__ hn,
                          const float* __restrict__ g, const float* __restrict__ be, int N) {
    int n = blockIdx.x;
    if (n >= N) return;
    int lane = threadIdx.x;
    float v0 = h[(long)n * H + lane]      + hn[(long)n * H + lane];
    float v1 = h[(long)n * H + lane + 32] + hn[(long)n * H + lane + 32];
    float s = v0 + v1;
#pragma unroll
    for (int m = 16; m >= 1; m >>= 1) s += __shfl_xor(s, m, 32);
    float mean = s * (1.0f / 64.0f);
    float d0 = v0 - mean, d1 = v1 - mean;
    float q = d0 * d0 + d1 * d1;
#pragma unroll
    for (int m = 16; m >= 1; m >>= 1) q += __shfl_xor(q, m, 32);
    float inv = rsqrtf(q * (1.0f / 64.0f) + 1e-5f);
    h[(long)n * H + lane]      = elu1(d0 * inv * g[lane]      + be[lane]);
    h[(long)n * H + lane + 32] = elu1(d1 * inv * g[lane + 32] + be[lane + 32]);
}

// ---------------- Cox head: hazard = elu(h@W1+b1)@W2 + b2 ----------------
__global__ void cox_kernel(const float* __restrict__ hp, const float* __restrict__ W1,
                           const float* __restrict__ b1, const float* __restrict__ W2,
                           const float* __restrict__ b2, float* __restrict__ out, int N) {
    int n = blockIdx.x * blockDim.x + threadIdx.x;
    if (n >= N) return;
    const float* hr = hp + (long)n * H;
    float hreg[H];
#pragma unroll
    for (int k = 0; k < H; ++k) hreg[k] = hr[k];
    float hz = b2[0];
    for (int j = 0; j < 32; ++j) {
        float s = b1[j];
#pragma unroll
        for (int k = 0; k < H; ++k) s = fmaf(hreg[k], W1[k * 32 + j], s);
        hz = fmaf(elu1(s), W2[j], hz);
    }
    out[n] = hz;
}

extern "C" void kernel_launch(void* const* d_in, const int* in_sizes, int n_in,
                              void* d_out, int out_size, void* d_ws, size_t ws_size,
                              hipStream_t stream) {
    (void)in_sizes; (void)n_in; (void)out_size; (void)ws_size;
    const int NG = 20000, NP = 100000, NM = 5000, E = 1000000;

    const float* x_gene    = (const float*)d_in[0];
    const float* x_patient = (const float*)d_in[1];
    const float* x_group   = (const float*)d_in[2];
    const int* srcA[4]  = { (const int*)d_in[3], (const int*)d_in[5],
                            (const int*)d_in[7], (const int*)d_in[9] };
    const int* dstA[4]  = { (const int*)d_in[4], (const int*)d_in[6],
                            (const int*)d_in[8], (const int*)d_in[10] };
    const float* encW[3] = { (const float*)d_in[11], (const float*)d_in[13], (const float*)d_in[15] };
    const float* encB[3] = { (const float*)d_in[12], (const float*)d_in[14], (const float*)d_in[16] };
    const float* convW_l = (const float*)d_in[17];
    const float* convb_l = (const float*)d_in[18];
    const float* convW_r = (const float*)d_in[19];
    const float* ln_g = (const float*)d_in[20];
    const float* ln_b = (const float*)d_in[21];
    const float* coxW1 = (const float*)d_in[22];
    const float* coxb1 = (const float*)d_in[23];
    const float* coxW2 = (const float*)d_in[24];
    const float* coxb2 = (const float*)d_in[25];

    float* ws = (float*)d_ws;
    float* h_gene  = ws; ws += (size_t)NG * H;
    float* h_pat   = ws; ws += (size_t)NP * H;
    float* h_grp   = ws; ws += (size_t)NM * H;
    float* hn_gene = ws; ws += (size_t)NG * H;
    float* hn_pat  = ws; ws += (size_t)NP * H;
    float* hn_grp  = ws; ws += (size_t)NM * H;
    float* agg     = ws; ws += (size_t)NP * H;   // reused for every edge type
    float* deg     = ws; ws += (size_t)NP;

    // node encoders
    enc_kernel<<<(NG * H + 255) / 256, 256, 0, stream>>>(x_gene,    encW[0], encB[0], h_gene, NG, 11);
    enc_kernel<<<(NP * H + 255) / 256, 256, 0, stream>>>(x_patient, encW[1], encB[1], h_pat,  NP, 3);
    enc_kernel<<<(NM * H + 255) / 256, 256, 0, stream>>>(x_group,   encW[2], encB[2], h_grp,  NM, 4);

    float* hArr[3]  = { h_gene, h_pat, h_grp };
    float* hnArr[3] = { hn_gene, hn_pat, hn_grp };
    int    nArr[3]  = { NG, NP, NM };
    const int sType[4] = { 0, 1, 0, 2 };   // gene, patient, gene, group
    const int dType[4] = { 1, 0, 2, 0 };   // patient, gene, group, gene

    for (int layer = 0; layer < 2; ++layer) {
        hipMemsetAsync(hn_gene, 0, (size_t)NG * H * sizeof(float), stream);
        hipMemsetAsync(hn_pat,  0, (size_t)NP * H * sizeof(float), stream);
        hipMemsetAsync(hn_grp,  0, (size_t)NM * H * sizeof(float), stream);
        for (int e = 0; e < 4; ++e) {
            int Nd = nArr[dType[e]];
            hipMemsetAsync(agg, 0, (size_t)Nd * H * sizeof(float), stream);
            hipMemsetAsync(deg, 0, (size_t)Nd * sizeof(float), stream);
            scatter_kernel<<<(E * 64) / 256, 256, 0, stream>>>(
                hArr[sType[e]], srcA[e], dstA[e], agg, deg, E);
            const float* Wl  = convW_l + ((size_t)layer * 4 + e) * H * H;
            const float* blp = convb_l + ((size_t)layer * 4 + e) * H;
            const float* Wr  = convW_r + ((size_t)layer * 4 + e) * H * H;
            sage_gemm_kernel<<<(Nd + 63) / 64, 128, 0, stream>>>(
                agg, deg, hArr[dType[e]], Wl, blp, Wr, hnArr[dType[e]], Nd);
        }
        for (int ni = 0; ni < 3; ++ni) {
            ln_kernel<<<nArr[ni], 32, 0, stream>>>(
                hArr[ni], hnArr[ni],
                ln_g + ((size_t)layer * 3 + ni) * H,
                ln_b + ((size_t)layer * 3 + ni) * H, nArr[ni]);
        }
    }
    cox_kernel<<<(NP + 255) / 256, 256, 0, stream>>>(
        h_pat, coxW1, coxb1, coxW2, coxb2, (float*)d_out, NP);
}